// LightGCN_24910810317344
// MI455X (gfx1250) — compile-verified
//
#include <hip/hip_runtime.h>

#define FEAT_D 128
#define SPMM_WAVES 4     // 4 wave32 per block -> 128 threads
#define SPMM_TILE 16     // neighbors staged per LDS tile (Poisson(10) degrees)

// ---------------- CDNA5 async global->LDS helpers ----------------
// GLOBAL_LOAD_ASYNC_TO_LDS_B128 (GV mode): per-lane 16B copy, tracked by
// ASYNCcnt. 32 lanes move one 512B feature row per instruction, no VGPR
// landing pad, deep outstanding-load window independent of LOADcnt.
static __device__ __forceinline__ void async_gather_b128(unsigned lds_off,
                                                         const void* gaddr) {
  unsigned long long ga = (unsigned long long)(size_t)gaddr;
  asm volatile("global_load_async_to_lds_b128 %0, %1, off"
               :: "v"(lds_off), "v"(ga)
               : "memory");
}

static __device__ __forceinline__ void wait_async0() {
  asm volatile("s_wait_asynccnt 0x0" ::: "memory");
}

// ---------------- CSR-build kernels (once per launch) ----------------
__global__ void hist_kernel(const int* __restrict__ rows,
                            int* __restrict__ deg, int nnz) {
  int i = blockIdx.x * blockDim.x + threadIdx.x;
  if (i < nnz) atomicAdd(&deg[rows[i]], 1);
}

// Allocate a contiguous range per row via a global ticket counter (avoids a scan).
__global__ void rowstart_kernel(const int* __restrict__ deg,
                                int* __restrict__ counter,
                                int* __restrict__ row_start,
                                int* __restrict__ cursor, int n) {
  int i = blockIdx.x * blockDim.x + threadIdx.x;
  if (i < n) {
    int d = deg[i];
    int s = atomicAdd(counter, d);
    row_start[i] = s;
    cursor[i] = s;
  }
}

// Pack {col, val} together: one 8B scattered store here, one 8B coalesced
// load on the SpMM dependent chain (instead of 2x 4B each).
__global__ void fill_kernel(const int* __restrict__ rows,
                            const int* __restrict__ cols,
                            const float* __restrict__ vals,
                            int* __restrict__ cursor,
                            int2* __restrict__ csr_pack, int nnz) {
  int i = blockIdx.x * blockDim.x + threadIdx.x;
  if (i < nnz) {
    int r = rows[i];
    int p = atomicAdd(&cursor[r], 1);
    csr_pack[p] = make_int2(cols[i], __float_as_int(vals[i]));
  }
}

// ---------------- fused SpMM layer ----------------
// One wave32 per row. Lane l owns feature columns [4l, 4l+4).
// Neighbor feature rows (512B each, L2-resident: whole matrix is 102.4MB of
// the 192MB L2) are pulled into LDS with async b128 copies, then reduced with
// packed FMAs. Writes x_out[row] (next layer input) and the accumulator:
//   acc_out[row] = (acc_in[row] + y[row]) * scale.
// Layer 1 passes acc_in = emb, acc_out = d_out (kills the copy pass);
// layers 2-3 pass acc_in = acc_out = d_out.
__global__ void spmm_csr_kernel(const float* __restrict__ x_in,
                                float* __restrict__ x_out,
                                const float* __restrict__ acc_in,
                                float* __restrict__ acc_out,
                                const int* __restrict__ row_start,
                                const int* __restrict__ deg,
                                const int2* __restrict__ csr_pack,
                                int n_rows, float scale, int write_x) {
  __shared__ float tile[SPMM_WAVES][SPMM_TILE][FEAT_D]; // 32 KB of 320 KB/WGP
  const int lane = threadIdx.x & 31;
  const int wave = threadIdx.x >> 5;
  const int row = blockIdx.x * SPMM_WAVES + wave;
  if (row >= n_rows) return;

  // generic LDS pointer: low 32 bits are the LDS byte offset (flat aperture rule)
  const unsigned lds_base = (unsigned)(size_t)(&tile[wave][0][0]);
  const size_t off = (size_t)row * FEAT_D + (size_t)(lane * 4);

  // Issue the independent accumulator load *before* the asm memory-clobber
  // barriers below; its latency hides under the whole gather/reduce phase.
  float4 g = *(const float4*)(acc_in + off);
  float4 a = make_float4(0.f, 0.f, 0.f, 0.f);

  const int base = row_start[row];
  const int dg = deg[row];

  for (int t = 0; t < dg; t += SPMM_TILE) {
    const int cnt = min(SPMM_TILE, dg - t);
    int c = 0;
    float v = 0.f;
    if (lane < cnt) {                       // coalesced b64 metadata load
      int2 m = csr_pack[base + t + lane];
      c = m.x;
      v = __int_as_float(m.y);
    }
    // Launch all gathers for this tile: one async b128 per neighbor (x32 lanes).
    for (int j = 0; j < cnt; ++j) {
      const int cj = __shfl(c, j, 32);
      async_gather_b128(lds_base + (unsigned)(j * (FEAT_D * 4)) +
                            (unsigned)(lane * 16),
                        x_in + (size_t)cj * FEAT_D + lane * 4);
    }
    wait_async0();
    // Reduce out of LDS.
    for (int j = 0; j < cnt; ++j) {
      const float vj = __shfl(v, j, 32);
      const float4 f = *(const float4*)&tile[wave][j][lane * 4];
      a.x = fmaf(vj, f.x, a.x);
      a.y = fmaf(vj, f.y, a.y);
      a.z = fmaf(vj, f.z, a.z);
      a.w = fmaf(vj, f.w, a.w);
    }
  }

  if (write_x) *(float4*)(x_out + off) = a;
  g.x = (g.x + a.x) * scale;
  g.y = (g.y + a.y) * scale;
  g.z = (g.z + a.z) * scale;
  g.w = (g.w + a.w) * scale;
  *(float4*)(acc_out + off) = g;
}

// ---------------- launch ----------------
extern "C" void kernel_launch(void* const* d_in, const int* in_sizes, int n_in,
                              void* d_out, int out_size, void* d_ws, size_t ws_size,
                              hipStream_t stream) {
  (void)n_in; (void)out_size; (void)ws_size;
  const float* emb  = (const float*)d_in[0];
  const float* vals = (const float*)d_in[1];
  const int*   rows = (const int*)d_in[2];
  const int*   cols = (const int*)d_in[3];
  const int ND  = in_sizes[0];          // N * 128
  const int n   = ND / FEAT_D;          // number of nodes
  const int nnz = in_sizes[1];          // symmetrized edge count

  // Workspace layout (512B aligned blocks):
  //   x0[ND] f32 | x1[ND] f32 | deg[n] counter[1] row_start[n] cursor[n]
  //   | csr_pack[nnz] int2
  char* ws = (char*)d_ws;
  const size_t xbytes = ((size_t)ND * sizeof(float) + 511) & ~(size_t)511;
  const size_t meta_bytes =
      (((size_t)(3 * n + 1) * sizeof(int)) + 511) & ~(size_t)511;
  float* x0        = (float*)(ws);
  float* x1        = (float*)(ws + xbytes);
  int*   deg       = (int*)(ws + 2 * xbytes);
  int*   counter   = deg + n;
  int*   row_start = counter + 1;
  int*   cursor    = row_start + n;
  int2*  csr_pack  = (int2*)(ws + 2 * xbytes + meta_bytes);

  float* acc = (float*)d_out;

  // zero degree histogram + ticket counter (graph-capture-safe)
  hipMemsetAsync(deg, 0, (size_t)(n + 1) * sizeof(int), stream);

  {
    int blk = 256, grd = (nnz + blk - 1) / blk;
    hist_kernel<<<grd, blk, 0, stream>>>(rows, deg, nnz);
  }
  {
    int blk = 256, grd = (n + blk - 1) / blk;
    rowstart_kernel<<<grd, blk, 0, stream>>>(deg, counter, row_start, cursor, n);
  }
  {
    int blk = 256, grd = (nnz + blk - 1) / blk;
    fill_kernel<<<grd, blk, 0, stream>>>(rows, cols, vals, cursor, csr_pack, nnz);
  }

  const int sgrd = (n + SPMM_WAVES - 1) / SPMM_WAVES;
  const int sblk = SPMM_WAVES * 32;
  // layer 1: y1 = A*emb ; acc = emb + y1 ; x0 = y1   (no separate copy pass)
  spmm_csr_kernel<<<sgrd, sblk, 0, stream>>>(emb, x0, emb, acc, row_start, deg,
                                             csr_pack, n, 1.0f, 1);
  // layer 2: y2 = A*x0 ; acc += y2 ; x1 = y2
  spmm_csr_kernel<<<sgrd, sblk, 0, stream>>>(x0, x1, acc, acc, row_start, deg,
                                             csr_pack, n, 1.0f, 1);
  // layer 3: y3 = A*x1 ; acc = (acc + y3) / 4   (mean over 4 terms)
  spmm_csr_kernel<<<sgrd, sblk, 0, stream>>>(x1, nullptr, acc, acc, row_start,
                                             deg, csr_pack, n, 0.25f, 0);
}